// SkipGram_60825326846574
// MI455X (gfx1250) — compile-verified
//
#include <hip/hip_runtime.h>

typedef __attribute__((ext_vector_type(2))) float v2f;
typedef __attribute__((ext_vector_type(8))) float v8f;

#define DIM 128
#define NEG 5
#define LMAX 4
#define NROWS 40960
#define ROWS_PER_BLOCK 16
#define LOCAL_ROWS 112      // 16 U rows + 16 V rows + 80 neg rows
#define LDS_STRIDE 132      // 128 floats + 4-float pad -> conflict-free column reads

__device__ __forceinline__ float log_sigmoid(float x) {
    // stable: min(x,0) - log1p(exp(-|x|))
    return fminf(x, 0.0f) - log1pf(__expf(-fabsf(x)));
}

__global__ void __launch_bounds__(256) skipgram_block_kernel(
    const float* __restrict__ u_table,
    const float* __restrict__ v_table,
    const int*   __restrict__ pos_u_idx,
    const int*   __restrict__ pos_u_len,
    const int*   __restrict__ pos_v_idx,
    const int*   __restrict__ pos_v_len,
    const int*   __restrict__ neg_v_idx,
    const int*   __restrict__ neg_v_len,
    float*       __restrict__ block_partials)
{
    __shared__ float lds[LOCAL_ROWS * LDS_STRIDE];
    __shared__ float wsum[8];

    const int tid  = threadIdx.x;
    const int wave = tid >> 5;
    const int lane = tid & 31;
    const int n0   = blockIdx.x * ROWS_PER_BLOCK;

    // ---------------- Phase 1: ragged bag-mean gathers -> LDS ----------------
    // 112 rows / 8 waves = 14 rows per wave; each lane covers 4 contiguous
    // floats (float4 -> global_load_b128, 512B coalesced per wave per row).
    // All 4 candidate gathers per bag are issued independently (slots beyond
    // `len` are clamped to slot 0 — duplicate addresses hit WGP$, costing no
    // HBM bandwidth) so 4 random-row gathers are in flight per iteration.
    for (int i = 0; i < 14; ++i) {
        const int lr = wave * 14 + i;
        const int* idxp;
        int len;
        const float* table;
        if (lr < 16) {
            idxp  = pos_u_idx + (size_t)(n0 + lr) * LMAX;
            len   = pos_u_len[n0 + lr];
            table = u_table;
        } else if (lr < 32) {
            idxp  = pos_v_idx + (size_t)(n0 + lr - 16) * LMAX;
            len   = pos_v_len[n0 + lr - 16];
            table = v_table;
        } else {
            const size_t j = (size_t)n0 * NEG + (lr - 32);
            idxp  = neg_v_idx + j * LMAX;
            len   = neg_v_len[j];
            table = v_table;
        }
        len = (len < 1) ? 1 : ((len > LMAX) ? LMAX : len);

        // one b128 load for all 4 bag indices
        const int4 iv = *reinterpret_cast<const int4*>(idxp);
        const size_t i0 = (size_t)iv.x;
        const size_t i1 = (size_t)((len > 1) ? iv.y : iv.x);
        const size_t i2 = (size_t)((len > 2) ? iv.z : iv.x);
        const size_t i3 = (size_t)((len > 3) ? iv.w : iv.x);

        const int col = lane * 4;
        const float4 e0 = *reinterpret_cast<const float4*>(table + i0 * DIM + col);
        const float4 e1 = *reinterpret_cast<const float4*>(table + i1 * DIM + col);
        const float4 e2 = *reinterpret_cast<const float4*>(table + i2 * DIM + col);
        const float4 e3 = *reinterpret_cast<const float4*>(table + i3 * DIM + col);

        const float f1 = (len > 1) ? 1.0f : 0.0f;
        const float f2 = (len > 2) ? 1.0f : 0.0f;
        const float f3 = (len > 3) ? 1.0f : 0.0f;
        const float inv = 1.0f / (float)len;

        float4 m;
        m.x = (e0.x + f1 * e1.x + f2 * e2.x + f3 * e3.x) * inv;
        m.y = (e0.y + f1 * e1.y + f2 * e2.y + f3 * e3.y) * inv;
        m.z = (e0.z + f1 * e1.z + f2 * e2.z + f3 * e3.z) * inv;
        m.w = (e0.w + f1 * e1.w + f2 * e2.w + f3 * e3.w) * inv;

        *reinterpret_cast<float4*>(&lds[lr * LDS_STRIDE + col]) = m;
    }

    __syncthreads();

    // ---------------- Phase 2: WMMA dot products + log-sigmoid ----------------
    // Wave 0: diag(U * V^T)  -> positive scores
    // Waves 1..5: diag(U * Neg_k^T) -> negative scores (k = wave-1)
    if (wave < 6) {
        const int r = lane & 15;   // matrix row handled by this lane
        const int h = lane >> 4;   // half-wave selects K pair {0,1} vs {2,3}

        const float* Arow = &lds[r * LDS_STRIDE];                       // U row r
        const float* Brow = (wave == 0)
            ? &lds[(16 + r) * LDS_STRIDE]                               // V row r
            : &lds[(32 + r * NEG + (wave - 1)) * LDS_STRIDE];           // Neg[r][k]

        v8f acc = {};
        #pragma unroll
        for (int k0 = 0; k0 < DIM; k0 += 4) {
            // f32 16x16x4 A layout: lane holds (row r, K = k0 + 2h + {0,1});
            // B (= V^T) layout mirrors it, so both feeds are 8B LDS loads.
            const float2 av = *reinterpret_cast<const float2*>(Arow + k0 + 2 * h);
            const float2 bv = *reinterpret_cast<const float2*>(Brow + k0 + 2 * h);
            v2f a; a.x = av.x; a.y = av.y;
            v2f b; b.x = bv.x; b.y = bv.y;
            acc = __builtin_amdgcn_wmma_f32_16x16x4_f32(
                false, a, false, b, (short)0, acc, false, false);
        }

        // Diagonal of the 16x16 f32 C/D layout:
        //   n in 0..7  -> vgpr n,   lane n      (lanes 0..7)
        //   n in 8..15 -> vgpr n-8, lane n+16   (lanes 24..31)
        int sel = -1;
        if (lane < 8)        sel = lane;
        else if (lane >= 24) sel = lane - 24;

        float d = 0.0f;
        #pragma unroll
        for (int i = 0; i < 8; ++i) {
            const float ci = acc[i];
            d = (sel == i) ? ci : d;
        }

        float contrib = 0.0f;
        if (sel >= 0) {
            const float x = (wave == 0) ? d : -d;   // ls(score) / ls(-neg_score)
            contrib = log_sigmoid(x);
        }

        // wave32 reduction
        #pragma unroll
        for (int off = 16; off >= 1; off >>= 1)
            contrib += __shfl_xor(contrib, off, 32);

        if (lane == 0) wsum[wave] = contrib;
    }

    __syncthreads();

    if (tid == 0) {
        float s = 0.0f;
        #pragma unroll
        for (int w = 0; w < 6; ++w) s += wsum[w];
        block_partials[blockIdx.x] = s;
    }
}

// Deterministic fixed-order final reduction (no float atomics).
__global__ void __launch_bounds__(256) skipgram_reduce_kernel(
    const float* __restrict__ partials,
    const int*   __restrict__ batch_ptr,
    float*       __restrict__ out,
    int n)
{
    __shared__ float sh[256];
    float s = 0.0f;
    for (int i = threadIdx.x; i < n; i += 256) s += partials[i];
    sh[threadIdx.x] = s;
    __syncthreads();
    #pragma unroll
    for (int k = 128; k >= 1; k >>= 1) {
        if (threadIdx.x < k) sh[threadIdx.x] += sh[threadIdx.x + k];
        __syncthreads();
    }
    if (threadIdx.x == 0)
        out[0] = -sh[0] / (float)batch_ptr[0];
}

extern "C" void kernel_launch(void* const* d_in, const int* in_sizes, int n_in,
                              void* d_out, int out_size, void* d_ws, size_t ws_size,
                              hipStream_t stream) {
    const float* u_table   = (const float*)d_in[0];
    const float* v_table   = (const float*)d_in[1];
    const int*   pos_u_idx = (const int*)d_in[2];
    const int*   pos_u_len = (const int*)d_in[3];
    const int*   pos_v_idx = (const int*)d_in[4];
    const int*   pos_v_len = (const int*)d_in[5];
    const int*   neg_v_idx = (const int*)d_in[6];
    const int*   neg_v_len = (const int*)d_in[7];
    const int*   batch_ptr = (const int*)d_in[8];
    float* out = (float*)d_out;
    float* ws  = (float*)d_ws;   // 2560 floats of scratch

    const int nblocks = NROWS / ROWS_PER_BLOCK;  // 2560

    hipLaunchKernelGGL(skipgram_block_kernel, dim3(nblocks), dim3(256), 0, stream,
                       u_table, v_table, pos_u_idx, pos_u_len,
                       pos_v_idx, pos_v_len, neg_v_idx, neg_v_len, ws);
    hipLaunchKernelGGL(skipgram_reduce_kernel, dim3(1), dim3(256), 0, stream,
                       ws, batch_ptr, out, nblocks);
}